// MDGAE_tfp2_65549790871682
// MI455X (gfx1250) — compile-verified
//
#include <hip/hip_runtime.h>
#include <hip/hip_bf16.h>
#include <math.h>

// ---------------------------------------------------------------------------
// Problem constants (match reference)
// ---------------------------------------------------------------------------
#define NN   100000
#define EE   1000000
#define DD   64
#define HH   14      // hidden width (padded to 16 for WMMA)
#define PP   16      // perturb width (already 16)
#define LATC 7
#define SP_INV_1f 0.5413248546129181f   // log(expm1(1))

typedef __attribute__((ext_vector_type(2))) float v2f;
typedef __attribute__((ext_vector_type(8))) float v8f;

// Packed-weight region layout (floats), appended after Ah/h1 in workspace:
//   W1pack : [   0, 1024)   16 k-steps x 32 lanes x 2
//   W2pack : [1024, 1280)    4 k-steps x 32 lanes x 2
//   Wd1pack: [1280, 1536)
//   Wd2pack: [1536, 1792)
//   bd1pad : [1792, 1808)
#define WPK_W1   0
#define WPK_W2   1024
#define WPK_WD1  1280
#define WPK_WD2  1536
#define WPK_BD1  1792
#define WPK_SIZE 1808

// ---------------------------------------------------------------------------
// Zero a float buffer
// ---------------------------------------------------------------------------
__global__ void mdgae_zero_kernel(float* __restrict__ p, int n) {
    int i = blockIdx.x * blockDim.x + threadIdx.x;
    if (i < n) p[i] = 0.0f;
}

// ---------------------------------------------------------------------------
// Pre-swizzle weights into the exact WMMA B-operand per-lane layout:
//   pack[ks*64 + L*2 + v] = Wpad[ ks*4 + 2*(L/16) + v ][ L%16 ]
// so each GEMM k-step becomes ONE unconditional b64 load per lane.
// Zero padding baked in here -> no EXEC-predicated loads in hot kernels.
// ---------------------------------------------------------------------------
__global__ __launch_bounds__(1024)
void mdgae_pack_kernel(const float* __restrict__ W1,   // 64 x 14
                       const float* __restrict__ W2,   // 14 x 14
                       const float* __restrict__ Wd1,  // 14 x 14
                       const float* __restrict__ Wd2,  // 14 x 16
                       const float* __restrict__ bd1,  // 14
                       float*       __restrict__ WP) {
    int t = threadIdx.x;                 // single block of 1024
    int ks = t >> 6;
    int r  = t & 63;
    int L  = r >> 1;
    int v  = r & 1;
    int kh = L >> 4;
    int n  = L & 15;
    int k  = ks * 4 + 2 * kh + v;

    // W1: 64 x 14, all k valid, pad n
    WP[WPK_W1 + t] = (n < HH) ? W1[k * HH + n] : 0.0f;

    if (t < 256) {   // ks in 0..3 here
        WP[WPK_W2  + t] = (k < HH && n < HH) ? W2 [k * HH + n] : 0.0f;
        WP[WPK_WD1 + t] = (k < HH && n < HH) ? Wd1[k * HH + n] : 0.0f;
        WP[WPK_WD2 + t] = (k < HH)           ? Wd2[k * PP + n] : 0.0f;
    }
    if (t < 16) {
        WP[WPK_BD1 + t] = (t < HH) ? bd1[t] : 0.0f;
    }
}

// ---------------------------------------------------------------------------
// SpMM #1: AX[rows[e], :] += vals[e] * X[cols[e], :]   (D = 64)
// 16 lanes per edge, each lane handles 4 consecutive floats (b128 gather).
// X (25.6 MB) and AX (25.6 MB) are L2-resident (192 MB L2).
// ---------------------------------------------------------------------------
__global__ void mdgae_spmm1_kernel(const float* __restrict__ X,
                                   const int*   __restrict__ rows,
                                   const int*   __restrict__ cols,
                                   const float* __restrict__ vals,
                                   float*       __restrict__ AX) {
    long long t = (long long)blockIdx.x * blockDim.x + threadIdx.x;
    int e  = (int)(t >> 4);
    int c4 = ((int)t & 15) * 4;
    if (e >= EE) return;
    int   cc = cols[e];
    int   rr = rows[e];
    float v  = vals[e];
    float4 x = *(const float4*)(X + (size_t)cc * DD + c4);
    float* op = AX + (size_t)rr * DD + c4;
    atomicAdd(op + 0, v * x.x);
    atomicAdd(op + 1, v * x.y);
    atomicAdd(op + 2, v * x.z);
    atomicAdd(op + 3, v * x.w);
}

// ---------------------------------------------------------------------------
// GEMM #1: h1 = relu(AX @ W1pad), one wave32 per 16-row tile.
// 16 x V_WMMA_F32_16X16X4_F32 over K=64; B from packed layout (1 b64/lane).
// ---------------------------------------------------------------------------
__global__ __launch_bounds__(256)
void mdgae_gemm1_kernel(const float* __restrict__ AX,
                        const float* __restrict__ WP,
                        float*       __restrict__ h1) {
    int wave = (int)((blockIdx.x * blockDim.x + threadIdx.x) >> 5);  // wave-uniform
    int lane = threadIdx.x & 31;
    if (wave >= NN / 16) return;   // uniform early-out: EXEC stays all-ones

    int m  = lane & 15;
    int kh = lane >> 4;
    int n  = m;
    size_t rowA = (size_t)(wave * 16 + m) * DD;
    const float* W1p = WP + WPK_W1;

    v8f c = {};
    #pragma unroll
    for (int ks = 0; ks < 16; ++ks) {
        int k0 = ks * 4 + 2 * kh;
        v2f a = *(const v2f*)(AX + rowA + k0);
        v2f b = *(const v2f*)(W1p + ks * 64 + lane * 2);
        c = __builtin_amdgcn_wmma_f32_16x16x4_f32(
                false, a, false, b, (short)0, c, false, false);
    }

    #pragma unroll
    for (int r = 0; r < 8; ++r) {
        int mrow = wave * 16 + r + 8 * kh;
        float v = c[r];
        h1[(size_t)mrow * 16 + n] = v > 0.0f ? v : 0.0f;
    }
}

// ---------------------------------------------------------------------------
// SpMM #2: Ah[rows[e], :] += vals[e] * h1[cols[e], :]   (padded width 16)
// 4 lanes per edge, 4 floats each. h1 (6.4 MB) is L2-resident.
// ---------------------------------------------------------------------------
__global__ void mdgae_spmm2_kernel(const float* __restrict__ h1,
                                   const int*   __restrict__ rows,
                                   const int*   __restrict__ cols,
                                   const float* __restrict__ vals,
                                   float*       __restrict__ Ah) {
    long long t = (long long)blockIdx.x * blockDim.x + threadIdx.x;
    int e  = (int)(t >> 2);
    int c4 = ((int)t & 3) * 4;
    if (e >= EE) return;
    int   cc = cols[e];
    int   rr = rows[e];
    float v  = vals[e];
    float4 x = *(const float4*)(h1 + (size_t)cc * 16 + c4);
    float* op = Ah + (size_t)rr * 16 + c4;
    atomicAdd(op + 0, v * x.x);
    atomicAdd(op + 1, v * x.y);
    atomicAdd(op + 2, v * x.z);
    atomicAdd(op + 3, v * x.w);
}

// ---------------------------------------------------------------------------
// Final fused kernel (one wave32 per 16-row tile, 2 waves/block, exact grid):
//   h  = relu(Ah @ W2pad)        (4 WMMA, K=16)
//   pd = tanh(h @ Wd1pad + bd1)  (4 WMMA)
//   pp = tanh(h @ Wd2pad + bd2)  (4 WMMA)
//   out[:, 0:7]=pd[:,0:7]; out[:,7:14]=softplus(pd[:,7:14]+c); out[:,14:30]=pp
// h bounced through LDS to convert C-layout -> A-layout.
// ---------------------------------------------------------------------------
__global__ __launch_bounds__(64)
void mdgae_final_kernel(const float* __restrict__ Ah,   // N x 16
                        const float* __restrict__ WP,   // packed weights
                        const float* __restrict__ bd2,  // 16
                        float*       __restrict__ out) { // N x 30
    __shared__ float lds[2][16 * 16];
    int wv   = threadIdx.x >> 5;
    int lane = threadIdx.x & 31;
    int tile = blockIdx.x * 2 + wv;     // grid exact: always < 6250

    int m  = lane & 15;
    int kh = lane >> 4;
    int n  = m;
    size_t baseRow = (size_t)tile * 16;

    const float* W2p  = WP + WPK_W2;
    const float* Wd1p = WP + WPK_WD1;
    const float* Wd2p = WP + WPK_WD2;

    // ---- h = relu(Ah @ W2pad) ----
    v8f ch = {};
    #pragma unroll
    for (int ks = 0; ks < 4; ++ks) {
        int k0 = ks * 4 + 2 * kh;
        v2f a = *(const v2f*)(Ah + (baseRow + m) * 16 + k0);
        v2f b = *(const v2f*)(W2p + ks * 64 + lane * 2);
        ch = __builtin_amdgcn_wmma_f32_16x16x4_f32(
                 false, a, false, b, (short)0, ch, false, false);
    }
    #pragma unroll
    for (int r = 0; r < 8; ++r) {
        float v = ch[r];
        lds[wv][(r + 8 * kh) * 16 + n] = v > 0.0f ? v : 0.0f;
    }
    __syncthreads();

    // Re-read h in A-operand layout
    v2f ha[4];
    #pragma unroll
    for (int ks = 0; ks < 4; ++ks) {
        int k0 = ks * 4 + 2 * kh;
        ha[ks] = *(const v2f*)(&lds[wv][m * 16 + k0]);
    }

    // ---- pd, pp GEMMs ----
    v8f cd = {}, cp = {};
    #pragma unroll
    for (int ks = 0; ks < 4; ++ks) {
        v2f b1 = *(const v2f*)(Wd1p + ks * 64 + lane * 2);
        v2f b2 = *(const v2f*)(Wd2p + ks * 64 + lane * 2);
        cd = __builtin_amdgcn_wmma_f32_16x16x4_f32(
                 false, ha[ks], false, b1, (short)0, cd, false, false);
        cp = __builtin_amdgcn_wmma_f32_16x16x4_f32(
                 false, ha[ks], false, b2, (short)0, cp, false, false);
    }

    float biasd = WP[WPK_BD1 + n];   // pre-padded, unconditional
    float biasp = bd2[n];

    #pragma unroll
    for (int r = 0; r < 8; ++r) {
        int mrow = tile * 16 + r + 8 * kh;
        float pd = tanhf(cd[r] + biasd);
        float pp = tanhf(cp[r] + biasp);
        if (n < LATC) {
            out[(size_t)mrow * 30 + n] = pd;
        } else if (n < HH) {
            float x  = pd + SP_INV_1f;
            float sp = (x > 0.0f) ? (x + log1pf(expf(-x))) : log1pf(expf(x));
            out[(size_t)mrow * 30 + n] = sp;
        }
        out[(size_t)mrow * 30 + 14 + n] = pp;
    }
}

// ---------------------------------------------------------------------------
// Launcher. Inputs: X, rows, cols, vals, W1, W2, Wd1, bd1, Wd2, bd2
// Workspace layout (floats):
//   AX : [0, N*64)             (25.6 MB) -- consumed by gemm1
//   h1 : [N*64, N*80)          ( 6.4 MB)
//   Ah : aliases [0, N*16)     (reuses AX region after gemm1 completes)
//   WP : [N*80, N*80 + 1808)   (~7 KB packed weights)
// ---------------------------------------------------------------------------
extern "C" void kernel_launch(void* const* d_in, const int* in_sizes, int n_in,
                              void* d_out, int out_size, void* d_ws, size_t ws_size,
                              hipStream_t stream) {
    const float* X    = (const float*)d_in[0];
    const int*   rows = (const int*)  d_in[1];
    const int*   cols = (const int*)  d_in[2];
    const float* vals = (const float*)d_in[3];
    const float* W1   = (const float*)d_in[4];
    const float* W2   = (const float*)d_in[5];
    const float* Wd1  = (const float*)d_in[6];
    const float* bd1  = (const float*)d_in[7];
    const float* Wd2  = (const float*)d_in[8];
    const float* bd2  = (const float*)d_in[9];
    float* out = (float*)d_out;

    float* ws = (float*)d_ws;
    float* AX = ws;                       // N*64 floats
    float* h1 = ws + (size_t)NN * 64;     // N*16 floats
    float* Ah = ws;                       // aliases AX region (N*16 floats)
    float* WP = ws + (size_t)NN * 80;     // packed weights

    // 0) pack weights into WMMA B-operand layout (zero-padded)
    mdgae_pack_kernel<<<1, 1024, 0, stream>>>(W1, W2, Wd1, Wd2, bd1, WP);
    // 1) zero AX
    {
        int n = NN * DD;
        mdgae_zero_kernel<<<(n + 255) / 256, 256, 0, stream>>>(AX, n);
    }
    // 2) SpMM #1 (atomic scatter-add)
    {
        long long t = (long long)EE * 16;
        mdgae_spmm1_kernel<<<(unsigned)((t + 255) / 256), 256, 0, stream>>>(
            X, rows, cols, vals, AX);
    }
    // 3) h1 = relu(AX @ W1)  [WMMA]
    {
        int waves  = NN / 16;                 // 6250
        int blocks = (waves + 7) / 8;         // 8 waves per 256-thread block
        mdgae_gemm1_kernel<<<blocks, 256, 0, stream>>>(AX, WP, h1);
    }
    // 4) zero Ah (reuses AX region)
    {
        int n = NN * 16;
        mdgae_zero_kernel<<<(n + 255) / 256, 256, 0, stream>>>(Ah, n);
    }
    // 5) SpMM #2
    {
        long long t = (long long)EE * 4;
        mdgae_spmm2_kernel<<<(unsigned)((t + 255) / 256), 256, 0, stream>>>(
            h1, rows, cols, vals, Ah);
    }
    // 6) fused tail MLP + epilogue  [WMMA]
    {
        int blocks = (NN / 16) / 2;           // 3125, exact
        mdgae_final_kernel<<<blocks, 64, 0, stream>>>(
            Ah, WP, bd2, out);
    }
}